// SelfAttentionLayers_65944927863373
// MI455X (gfx1250) — compile-verified
//
#include <hip/hip_runtime.h>
#include <math.h>

typedef float v2f __attribute__((ext_vector_type(2)));
typedef float v8f __attribute__((ext_vector_type(8)));

#define S 512
#define B 64
#define D 1024
#define H 2048
#define NH 8
#define M_TOT (S*B)          // 32768 flattened (s,b) rows, D contiguous
#define MT 64                // rows per workgroup
#define LDX 1028             // padded LDS stride for X tile (floats); 1028%64==4 -> 16 rows hit 16 banks
#define KC 32                // K chunk for W staging
#define NTH 128              // h tile per outer iteration (8 waves = 2M x 4N of 32x32 blocks)
#define LDW 36               // padded LDS stride for W tile rows (36: gcd(36,64)=4, 9 coprime 16 -> conflict free)
#define LDA 132              // padded stride for activated tile / W_att tile (132%4==0 -> float4 aligned)

#define SX_SZ (MT*LDX)       // 65792 floats
#define SW_SZ (NTH*LDW)      // 4608
#define SACT_SZ (MT*LDA)     // 8448
#define SWATT_SZ (NH*LDA)    // 1056
#define SMEM_FLOATS (SX_SZ+SW_SZ+SACT_SZ+SWATT_SZ)  // 79904 floats = 319,616 B <= 320 KB

// ---------------------------------------------------------------------------
// Kernel A: fused  logits[m,a] = relu(X[m,:]·W_red^T + b_red)·W_att^T + b_att
// One workgroup = 64 rows.  8 waves = 2 (M, 32 rows) x 4 (N, 32 h);
// each wave holds a 2x2 block of 16x16 f32 WMMA accumulators (register
// blocking: 4 LDS operand loads feed 4 WMMAs -> 256 B LDS / WMMA, balancing
// LDS bandwidth against the matrix pipe).  X tile LDS-resident (full K);
// W_red streamed in 32-wide K chunks.  V_WMMA_F32_16X16X4_F32 layouts:
//   A 16x4 f32 : lane&15=M, VGPR v holds K = v + 2*(lane>>4)
//   B  4x16 f32: lane&15=N, VGPR v holds K = v + 2*(lane>>4)
//   C/D 16x16  : lane&15=N, VGPR v holds M = v + 8*(lane>>4)
// ---------------------------------------------------------------------------
__global__ void __launch_bounds__(256)
fused_logits_kernel(const float* __restrict__ X, const float* __restrict__ Wred,
                    const float* __restrict__ bred, const float* __restrict__ Watt,
                    const float* __restrict__ batt, float* __restrict__ logits)
{
  extern __shared__ float smem[];
  float* sX    = smem;
  float* sW    = sX + SX_SZ;
  float* sAct  = sW + SW_SZ;
  float* sWatt = sAct + SACT_SZ;

  const int tid   = threadIdx.x;
  const int m0    = blockIdx.x * MT;
  const int lane  = tid & 31;
  const int wid   = tid >> 5;
  const int waveM = wid & 1;    // 32-row strip
  const int waveN = wid >> 1;   // 32-h strip (of the 128-h tile)
  const int lm    = lane & 15;
  const int lh    = lane >> 4;  // half-wave selector -> K offset 0/2 (A,B), M offset 0/8 (C)

  // ---- stage X tile [MT x D] into LDS (coalesced float4) ----
  {
    const float4* Xg = (const float4*)(X + (size_t)m0 * D);
    for (int slot = tid; slot < MT * (D/4); slot += 256) {
      int row = slot >> 8;        // D/4 == 256 float4 per row
      int c4  = slot & 255;
      float4 v = Xg[row * (D/4) + c4];
      float* dst = sX + row * LDX + c4 * 4;
      dst[0]=v.x; dst[1]=v.y; dst[2]=v.z; dst[3]=v.w;
    }
  }

  float lacc0 = 0.f, lacc1 = 0.f;   // per-thread logit accumulators (2 of 512 (m,a) pairs)

  const int aBase0 = (waveM*32 + lm) * LDX + 2*lh;   // M-sub 0
  const int aBase1 = aBase0 + 16 * LDX;              // M-sub 1
  const int bBase0 = (waveN*32 + lm) * LDW + 2*lh;   // N-sub 0
  const int bBase1 = bBase0 + 16 * LDW;              // N-sub 1

  for (int nt = 0; nt < H/NTH; ++nt) {
    const int h0 = nt * NTH;
    __syncthreads();  // covers initial X stage + smem reuse from previous iteration

    // stage W_att tile 8 x 128 (4 floats per thread)
    for (int slot = tid; slot < NH * NTH; slot += 256) {
      int a = slot >> 7, hh = slot & 127;
      sWatt[a * LDA + hh] = Watt[(size_t)a * H + h0 + hh];
    }

    v8f acc00 = {0.f,0.f,0.f,0.f,0.f,0.f,0.f,0.f};
    v8f acc01 = acc00, acc10 = acc00, acc11 = acc00;

    for (int kc = 0; kc < D; kc += KC) {
      // stage W_red chunk: 128 h-rows x KC cols (float4; 4 per thread)
      for (int slot = tid; slot < NTH * (KC/4); slot += 256) {
        int row = slot >> 3;      // KC/4 == 8 float4 per row
        int c4  = slot & 7;
        const float4* wg = (const float4*)(Wred + (size_t)(h0 + row) * D + kc);
        float4 v = wg[c4];
        float* dst = sW + row * LDW + c4 * 4;
        dst[0]=v.x; dst[1]=v.y; dst[2]=v.z; dst[3]=v.w;
      }
      __syncthreads();
      #pragma unroll
      for (int k = 0; k < KC; k += 4) {
        v2f a0, a1, b0, b1;
        a0.x = sX[aBase0 + kc + k];  a0.y = sX[aBase0 + kc + k + 1];
        a1.x = sX[aBase1 + kc + k];  a1.y = sX[aBase1 + kc + k + 1];
        b0.x = sW[bBase0 + k];       b0.y = sW[bBase0 + k + 1];
        b1.x = sW[bBase1 + k];       b1.y = sW[bBase1 + k + 1];
        acc00 = __builtin_amdgcn_wmma_f32_16x16x4_f32(false, a0, false, b0, (short)0, acc00, false, false);
        acc01 = __builtin_amdgcn_wmma_f32_16x16x4_f32(false, a0, false, b1, (short)0, acc01, false, false);
        acc10 = __builtin_amdgcn_wmma_f32_16x16x4_f32(false, a1, false, b0, (short)0, acc10, false, false);
        acc11 = __builtin_amdgcn_wmma_f32_16x16x4_f32(false, a1, false, b1, (short)0, acc11, false, false);
      }
      __syncthreads();  // protect sW before restaging
    }

    // bias + relu -> sAct[64 m][128 h]
    {
      const int hl0 = waveN*32 + lm;        // N-sub 0 column
      const int hl1 = hl0 + 16;             // N-sub 1 column
      const float bias0 = bred[h0 + hl0];
      const float bias1 = bred[h0 + hl1];
      #pragma unroll
      for (int v = 0; v < 8; ++v) {
        int ml0 = waveM*32 + v + 8*lh;      // M-sub 0 row
        int ml1 = ml0 + 16;                 // M-sub 1 row
        float v00 = acc00[v] + bias0, v01 = acc01[v] + bias1;
        float v10 = acc10[v] + bias0, v11 = acc11[v] + bias1;
        sAct[ml0 * LDA + hl0] = v00 > 0.f ? v00 : 0.f;
        sAct[ml0 * LDA + hl1] = v01 > 0.f ? v01 : 0.f;
        sAct[ml1 * LDA + hl0] = v10 > 0.f ? v10 : 0.f;
        sAct[ml1 * LDA + hl1] = v11 > 0.f ? v11 : 0.f;
      }
    }
    __syncthreads();

    // fold this 128-h slab into logits: 512 (m,a) pairs, 2 per thread, float4 reads
    {
      const int ma = tid >> 3, aa = tid & 7;
      const int mb = (tid + 256) >> 3;      // second pair has same a
      const float4* actA = (const float4*)(sAct + ma * LDA);
      const float4* actB = (const float4*)(sAct + mb * LDA);
      const float4* wv   = (const float4*)(sWatt + aa * LDA);
      float s0 = 0.f, s1 = 0.f;
      #pragma unroll 8
      for (int q = 0; q < NTH/4; ++q) {
        float4 w  = wv[q];
        float4 x0 = actA[q];
        float4 x1 = actB[q];
        s0 = fmaf(x0.x, w.x, fmaf(x0.y, w.y, fmaf(x0.z, w.z, fmaf(x0.w, w.w, s0))));
        s1 = fmaf(x1.x, w.x, fmaf(x1.y, w.y, fmaf(x1.z, w.z, fmaf(x1.w, w.w, s1))));
      }
      lacc0 += s0; lacc1 += s1;
    }
  }

  { // write logits (+ b_att), coalesced
    const int ma = tid >> 3,       aa = tid & 7;
    const int mb = (tid+256) >> 3;
    logits[(size_t)(m0 + ma) * NH + aa] = lacc0 + batt[aa];
    logits[(size_t)(m0 + mb) * NH + aa] = lacc1 + batt[aa];
  }
}

// ---------------------------------------------------------------------------
// Kernel B: softmax over S (axis 0), in place. One block per (b,a) column.
// ---------------------------------------------------------------------------
__global__ void __launch_bounds__(256)
softmax_kernel(float* __restrict__ attn)
{
  __shared__ float red[256];
  const int col = blockIdx.x;                 // b*NH + a
  const int tid = threadIdx.x;
  float v0 = attn[(size_t)tid * (B*NH) + col];
  float v1 = attn[(size_t)(tid + 256) * (B*NH) + col];
  red[tid] = fmaxf(v0, v1); __syncthreads();
  for (int off = 128; off > 0; off >>= 1) {
    if (tid < off) red[tid] = fmaxf(red[tid], red[tid + off]);
    __syncthreads();
  }
  float mx = red[0]; __syncthreads();
  float e0 = expf(v0 - mx), e1 = expf(v1 - mx);
  red[tid] = e0 + e1; __syncthreads();
  for (int off = 128; off > 0; off >>= 1) {
    if (tid < off) red[tid] += red[tid + off];
    __syncthreads();
  }
  float inv = 1.0f / red[0];
  attn[(size_t)tid * (B*NH) + col]         = e0 * inv;
  attn[(size_t)(tid + 256) * (B*NH) + col] = e1 * inv;
}

// ---------------------------------------------------------------------------
// Kernel C: penalty per batch via Gram trick:
//   ||A A^T - I||_F^2 = ||G||_F^2 - 2 tr(G) + S,  G = A^T A  (8x8)
// ---------------------------------------------------------------------------
__global__ void __launch_bounds__(256)
penalty_kernel(const float* __restrict__ attn, float* __restrict__ penb)
{
  __shared__ float sg[36];
  const int b = blockIdx.x, tid = threadIdx.x;
  float g[36];
  #pragma unroll
  for (int i = 0; i < 36; ++i) g[i] = 0.f;
  for (int s = tid; s < S; s += 256) {
    const float* p = attn + (size_t)s * (B*NH) + b * NH;
    float v[8];
    #pragma unroll
    for (int a = 0; a < 8; ++a) v[a] = p[a];
    int c = 0;
    #pragma unroll
    for (int i = 0; i < 8; ++i)
      #pragma unroll
      for (int j = i; j < 8; ++j) { g[c] = fmaf(v[i], v[j], g[c]); ++c; }
  }
  if (tid < 36) sg[tid] = 0.f;
  __syncthreads();
  #pragma unroll
  for (int i = 0; i < 36; ++i) atomicAdd(&sg[i], g[i]);
  __syncthreads();
  if (tid == 0) {
    float fro = 0.f, tr = 0.f; int c = 0;
    for (int i = 0; i < 8; ++i)
      for (int j = i; j < 8; ++j) {
        float x = sg[c++];
        fro += (i == j) ? x*x : 2.f*x*x;
        if (i == j) tr += x;
      }
    penb[b] = sqrtf(fmaxf(fro - 2.f*tr + (float)S, 0.f));
  }
}

__global__ void penalty_sum_kernel(const float* __restrict__ penb, float* __restrict__ out_pen)
{
  if (threadIdx.x == 0) {
    float s = 0.f;
    for (int b = 0; b < B; ++b) s += penb[b];
    *out_pen = s;
  }
}

// ---------------------------------------------------------------------------
// Kernel D: attended[b, a, d] = sum_s attn[s,b,a] * X[s,b,d]   (memory-bound)
// Grid (D/256, B); attn column staged in LDS, X read coalesced once.
// ---------------------------------------------------------------------------
__global__ void __launch_bounds__(256)
attended_kernel(const float* __restrict__ attn, const float* __restrict__ X,
                float* __restrict__ out)
{
  __shared__ float sA[S*NH];                     // 16 KB
  const int b = blockIdx.y;
  const int d = blockIdx.x * 256 + threadIdx.x;
  for (int idx = threadIdx.x; idx < S*NH; idx += 256) {
    int s = idx >> 3, a = idx & 7;
    sA[idx] = attn[(size_t)s * (B*NH) + b*NH + a];
  }
  __syncthreads();
  float acc[8];
  #pragma unroll
  for (int a = 0; a < 8; ++a) acc[a] = 0.f;
  for (int s = 0; s < S; ++s) {
    float x = X[((size_t)s * B + b) * D + d];
    #pragma unroll
    for (int a = 0; a < 8; ++a) acc[a] = fmaf(sA[s*8 + a], x, acc[a]);
  }
  #pragma unroll
  for (int a = 0; a < 8; ++a)
    out[(size_t)b * (NH*D) + a*D + d] = acc[a];
}

// ---------------------------------------------------------------------------
extern "C" void kernel_launch(void* const* d_in, const int* in_sizes, int n_in,
                              void* d_out, int out_size, void* d_ws, size_t ws_size,
                              hipStream_t stream)
{
  (void)in_sizes; (void)n_in; (void)out_size; (void)ws_size;
  const float* X    = (const float*)d_in[0];   // [S,B,D]
  const float* Wred = (const float*)d_in[1];   // [H,D]
  const float* bred = (const float*)d_in[2];   // [H]
  const float* Watt = (const float*)d_in[3];   // [NH,H]
  const float* batt = (const float*)d_in[4];   // [NH]
  float* out    = (float*)d_out;               // attended [B, NH*D] then penalty scalar
  float* logits = (float*)d_ws;                // S*B*NH floats (reused as attn in-place)
  float* penb   = logits + (size_t)S*B*NH;     // B floats

  const size_t smemA = (size_t)SMEM_FLOATS * sizeof(float);
  fused_logits_kernel<<<M_TOT/MT, 256, smemA, stream>>>(X, Wred, bred, Watt, batt, logits);
  softmax_kernel<<<B*NH, 256, 0, stream>>>(logits);
  penalty_kernel<<<B, 256, 0, stream>>>(logits, penb);
  penalty_sum_kernel<<<1, 32, 0, stream>>>(penb, out + (size_t)B*NH*D);
  attended_kernel<<<dim3(D/256, B), 256, 0, stream>>>(logits, X, out);
}